// MinusAtten_62002147885097
// MI455X (gfx1250) — compile-verified
//
#include <hip/hip_runtime.h>
#include <cstddef>

#define BATCH 2
#define S_LEN 512
#define D_DIM 256

typedef __attribute__((ext_vector_type(2))) float v2f;
typedef __attribute__((ext_vector_type(4))) float v4f;
typedef __attribute__((ext_vector_type(8))) float v8f;

// ---------- fast transcendental helpers (gfx1250 TRANS ops) ----------
__device__ __forceinline__ float fast_exp2(float x) {
#if __has_builtin(__builtin_amdgcn_exp2f)
    return __builtin_amdgcn_exp2f(x);
#else
    return exp2f(x);
#endif
}

__device__ __forceinline__ float fast_rcp(float x) {
#if __has_builtin(__builtin_amdgcn_rcpf)
    return __builtin_amdgcn_rcpf(x);
#else
    return 1.0f / x;
#endif
}

__device__ __forceinline__ float fast_tanh(float x) {
#if __has_builtin(__builtin_amdgcn_tanhf)
    return __builtin_amdgcn_tanhf(x);          // v_tanh_f32
#else
    // tanh(x) = sign(x) * (1 - e^{-2|x|}) / (1 + e^{-2|x|}), e^{-2|x|} via exp2
    float ax = fabsf(x);
    float t  = fast_exp2(-2.8853900817779268f * ax);   // 2*log2(e)
    float r  = (1.0f - t) * fast_rcp(1.0f + t);
    return copysignf(r, x);
#endif
}

// ---------------------------------------------------------------------
// Kernel 1: P[m, e] = sum_d q[m, d] * Wm[e, d]   (M = B*S = 1024, K = N = 256)
// One wave per 16x16 tile of P, fp32 WMMA 16x16x4, 64 k-steps.
// A (16x4 f32) per-lane: lane<16 -> K={0,1}, lane>=16 -> K={2,3}; M = lane&15.
// B (4x16 f32) per-lane: same K split; N = lane&15.  B[k][n] = Wm[n][k].
// C (16x16 f32): 8 VGPRs; N = lane&15; M = r (+8 for lanes 16-31).
// ---------------------------------------------------------------------
__global__ __launch_bounds__(256) void proj_gemm_wmma(
    const float* __restrict__ q,   // [B*S, D]
    const float* __restrict__ Wm,  // [D, D] row-major (e, d)
    float* __restrict__ P)         // [B*S, D]
{
    const int lane = threadIdx.x & 31;
    const int wave = blockIdx.x * (blockDim.x >> 5) + (threadIdx.x >> 5);

    const int NT   = D_DIM / 16;        // 16 tiles along N
    const int tm   = wave / NT;         // 0..63
    const int tn   = wave % NT;         // 0..15

    const int half = lane >> 4;         // 0 or 1
    const int l15  = lane & 15;
    const int koff = half * 2;

    const float* qrow = q  + (size_t)(tm * 16 + l15) * D_DIM;
    const float* wrow = Wm + (size_t)(tn * 16 + l15) * D_DIM;

    v8f acc = {};
    for (int k = 0; k < D_DIM; k += 4) {
        v2f a, b;
        a.x = qrow[k + koff];
        a.y = qrow[k + koff + 1];
        b.x = wrow[k + koff];
        b.y = wrow[k + koff + 1];
        acc = __builtin_amdgcn_wmma_f32_16x16x4_f32(
            /*neg_a=*/false, a, /*neg_b=*/false, b,
            /*c_mod=*/(short)0, acc, /*reuse_a=*/false, /*reuse_b=*/false);
    }

    float* pout = P + (size_t)(tm * 16 + half * 8) * D_DIM + tn * 16 + l15;
#pragma unroll
    for (int r = 0; r < 8; ++r)
        pout[(size_t)r * D_DIM] = acc[r];
}

// ---------------------------------------------------------------------
// Kernel 2: one block per attention row (b, i).
//   scores:  s_j = sum_e v[e] * tanh(P[b,j,e] - P[b,i,e])   (8 waves x 64 j)
//            lane owns e in [lane*8, lane*8+8) -> 2 x b128 loads per j
//   softmax over j, write atten row
//   context: ctx[d] = sum_j w_j * value[b,j,d]              (thread <-> d)
// ---------------------------------------------------------------------
__global__ __launch_bounds__(256) void atten_rows(
    const float* __restrict__ P,      // [B*S, D]
    const float* __restrict__ vmw,    // [D]
    const float* __restrict__ value,  // [B*S, D]
    float* __restrict__ ctx_out,      // [B*S, D]
    float* __restrict__ atten_out)    // [B*S, S]
{
    __shared__ float s_w[S_LEN];      // scores, then softmax weights
    __shared__ float s_red[256];

    const int t    = threadIdx.x;
    const int lane = t & 31;
    const int wv   = t >> 5;          // wave id: 0..7
    const int row  = blockIdx.x;      // b*S + i
    const int b    = row / S_LEN;

    // register-resident per-lane slices of P_i and vm: e = lane*8 + k
    const v4f* Pi4 = (const v4f*)(P + (size_t)row * D_DIM + lane * 8);
    const v4f* vm4 = (const v4f*)(vmw + lane * 8);
    v4f pi0 = Pi4[0], pi1 = Pi4[1];
    v4f vv0 = vm4[0], vv1 = vm4[1];

    // ---- scores: each wave handles 64 consecutive j ----
    const float* Pb = P + (size_t)b * S_LEN * D_DIM + lane * 8;
    for (int jj = 0; jj < 64; ++jj) {
        const int j = wv * 64 + jj;
        const v4f* Pj4 = (const v4f*)(Pb + (size_t)j * D_DIM);
        v4f a0 = Pj4[0];               // wave reads one contiguous 1KB row
        v4f a1 = Pj4[1];
        float partial = 0.0f;
#pragma unroll
        for (int k = 0; k < 4; ++k) {
            partial = fmaf(vv0[k], fast_tanh(a0[k] - pi0[k]), partial);
            partial = fmaf(vv1[k], fast_tanh(a1[k] - pi1[k]), partial);
        }
#pragma unroll
        for (int off = 16; off > 0; off >>= 1)
            partial += __shfl_xor(partial, off, 32);
        if (lane == 0) s_w[j] = partial;
    }
    __syncthreads();

    // ---- softmax over 512 scores (2 per thread) ----
    float s0 = s_w[t];
    float s1 = s_w[t + 256];

    s_red[t] = fmaxf(s0, s1);
    __syncthreads();
    for (int off = 128; off > 0; off >>= 1) {
        if (t < off) s_red[t] = fmaxf(s_red[t], s_red[t + off]);
        __syncthreads();
    }
    const float mx = s_red[0];
    __syncthreads();

    const float L2E = 1.4426950408889634f;
    float e0 = fast_exp2((s0 - mx) * L2E);
    float e1 = fast_exp2((s1 - mx) * L2E);

    s_red[t] = e0 + e1;
    __syncthreads();
    for (int off = 128; off > 0; off >>= 1) {
        if (t < off) s_red[t] += s_red[t + off];
        __syncthreads();
    }
    const float inv = fast_rcp(s_red[0]);

    const float w0 = e0 * inv;
    const float w1 = e1 * inv;
    s_w[t]       = w0;
    s_w[t + 256] = w1;

    float* arow = atten_out + (size_t)row * S_LEN;
    arow[t]       = w0;
    arow[t + 256] = w1;
    __syncthreads();

    // ---- context row: thread t owns d = t, coalesced value loads ----
    const float* Vb = value + (size_t)b * S_LEN * D_DIM + t;
    float acc0 = 0.0f, acc1 = 0.0f, acc2 = 0.0f, acc3 = 0.0f;
    for (int j = 0; j < S_LEN; j += 4) {
        acc0 = fmaf(s_w[j + 0], Vb[(size_t)(j + 0) * D_DIM], acc0);
        acc1 = fmaf(s_w[j + 1], Vb[(size_t)(j + 1) * D_DIM], acc1);
        acc2 = fmaf(s_w[j + 2], Vb[(size_t)(j + 2) * D_DIM], acc2);
        acc3 = fmaf(s_w[j + 3], Vb[(size_t)(j + 3) * D_DIM], acc3);
    }
    ctx_out[(size_t)row * D_DIM + t] = (acc0 + acc1) + (acc2 + acc3);
}

// ---------------------------------------------------------------------
extern "C" void kernel_launch(void* const* d_in, const int* in_sizes, int n_in,
                              void* d_out, int out_size, void* d_ws, size_t ws_size,
                              hipStream_t stream) {
    (void)in_sizes; (void)n_in; (void)out_size; (void)ws_size;

    const float* q     = (const float*)d_in[0];  // query [B,S,D]
    // d_in[1] = key: unused by the reference math
    const float* value = (const float*)d_in[2];  // [B,S,D]
    const float* Wm    = (const float*)d_in[3];  // [D,D]
    const float* vmw   = (const float*)d_in[4];  // [1,D]

    float* ctx   = (float*)d_out;                                 // [B,S,D]
    float* atten = (float*)d_out + (size_t)BATCH * S_LEN * D_DIM; // [B,S,S]
    float* P     = (float*)d_ws;                                  // [B*S, D] = 1 MB

    // Kernel 1: 1024 16x16 tiles, 1 wave each, 8 waves/block -> 128 blocks
    const int waves = (BATCH * S_LEN / 16) * (D_DIM / 16);
    proj_gemm_wmma<<<dim3(waves / 8), dim3(256), 0, stream>>>(q, Wm, P);

    // Kernel 2: one block per (b, i) row
    atten_rows<<<dim3(BATCH * S_LEN), dim3(256), 0, stream>>>(P, vmw, value, ctx, atten);
}